// HexPlaneField_t_84937273245763
// MI455X (gfx1250) — compile-verified
//
#include <hip/hip_runtime.h>
#include <stdint.h>

// HexPlane field sampling, MI455X (gfx1250, wave32).
// - 4 points/wave, 8 lanes/point, 4 channels/lane -> all corner gathers are
//   global_load_b128 (512 B of useful data per vmem instruction).
// - Plane working set ~56 MB -> L2-resident (192 MB L2); output stream (192 MB)
//   written with non-temporal b128 stores so it cannot evict the planes.
// - Per-level quantization hoisted: only cx/cy/cz @ S and ct @ 150 are
//   quantized (4 per level instead of 12), row/col offset terms precomputed.
// - Point coordinates staged wave-locally through LDS with
//   global_load_async_to_lds_b32 + s_wait_asynccnt (ASYNCcnt path).

typedef float v4f __attribute__((ext_vector_type(4)));

struct PlanePtrs {
    const float* p[18];   // [level][comb] flattened, level-major
};

struct Q {
    int   q0, q1;   // clamped grid indices
    float w0, w1;   // (1-frac), frac
};

template <int R>
__device__ __forceinline__ Q quant(float c) {
    float f = (c + 1.0f) * (0.5f * (float)(R - 1));
    f = fminf(fmaxf(f, 0.0f), (float)(R - 1));
    const int i0 = (int)f;                       // f >= 0 -> trunc == floor
    const int i1 = (i0 + 1 < R) ? i0 + 1 : R - 1;
    Q q;
    q.q0 = i0;
    q.q1 = i1;
    q.w1 = f - (float)i0;
    q.w0 = 1.0f - q.w1;
    return q;
}

// One bilinear sample from precomputed row/col offset terms and weights.
__device__ __forceinline__ void sample2(const float* __restrict__ base,
                                        int r0, int r1, float wy0, float wy1,
                                        int c0, int c1, float wx0, float wx1,
                                        v4f& acc) {
    const v4f v00 = *(const v4f*)(base + r0 + c0);
    const v4f v01 = *(const v4f*)(base + r0 + c1);
    const v4f v10 = *(const v4f*)(base + r1 + c0);
    const v4f v11 = *(const v4f*)(base + r1 + c1);
    acc += v00 * (wx0 * wy0) + v01 * (wx1 * wy0) +
           v10 * (wx0 * wy1) + v11 * (wx1 * wy1);
}

// Sum of the 6 coordinate-pair planes at one multires level.
// COMBS = (0,1),(0,2),(0,3),(1,2),(1,3),(2,3); plane(i,j): W=reso[i], H=reso[j].
// i is always spatial (W = S); row stride is always S*32 floats.
template <int LVL>
__device__ __forceinline__ v4f level_feat(const PlanePtrs& pl,
                                          float cx, float cy, float cz,
                                          const Q& qt, int cg4) {
    constexpr int M  = (LVL == 0) ? 1 : (LVL == 1) ? 2 : 4;
    constexpr int S  = 64 * M;
    constexpr int RS = S * 32;        // row stride in floats

    const Q qx = quant<S>(cx);
    const Q qy = quant<S>(cy);
    const Q qz = quant<S>(cz);

    // column terms (coordinate used on the W axis)
    const int xc0 = qx.q0 * 32 + cg4, xc1 = qx.q1 * 32 + cg4;
    const int yc0 = qy.q0 * 32 + cg4, yc1 = qy.q1 * 32 + cg4;
    const int zc0 = qz.q0 * 32 + cg4, zc1 = qz.q1 * 32 + cg4;
    // row terms (coordinate used on the H axis)
    const int yr0 = qy.q0 * RS, yr1 = qy.q1 * RS;
    const int zr0 = qz.q0 * RS, zr1 = qz.q1 * RS;
    const int tr0 = qt.q0 * RS, tr1 = qt.q1 * RS;

    v4f acc = {0.0f, 0.0f, 0.0f, 0.0f};
    sample2(pl.p[LVL * 6 + 0], yr0, yr1, qy.w0, qy.w1, xc0, xc1, qx.w0, qx.w1, acc); // (x,y)
    sample2(pl.p[LVL * 6 + 1], zr0, zr1, qz.w0, qz.w1, xc0, xc1, qx.w0, qx.w1, acc); // (x,z)
    sample2(pl.p[LVL * 6 + 2], tr0, tr1, qt.w0, qt.w1, xc0, xc1, qx.w0, qx.w1, acc); // (x,t)
    sample2(pl.p[LVL * 6 + 3], zr0, zr1, qz.w0, qz.w1, yc0, yc1, qy.w0, qy.w1, acc); // (y,z)
    sample2(pl.p[LVL * 6 + 4], tr0, tr1, qt.w0, qt.w1, yc0, yc1, qy.w0, qy.w1, acc); // (y,t)
    sample2(pl.p[LVL * 6 + 5], tr0, tr1, qt.w0, qt.w1, zc0, zc1, qz.w0, qz.w1, acc); // (z,t)
    return acc;
}

__global__ __launch_bounds__(256) void hexplane_kernel(
    const float* __restrict__ pts,        // [N,3]
    const float* __restrict__ tsv,        // [N,1]
    PlanePtrs pl,
    float* __restrict__ out,              // [N,96]
    int n) {
    // Per-wave coord staging: 16 floats per wave (12 xyz + 4 t), 8 waves/block.
    __shared__ float csmem[8 * 16];

    const int lane  = threadIdx.x & 31;
    const int wslot = threadIdx.x >> 5;                             // wave in block
    const int wid   = (blockIdx.x * blockDim.x + threadIdx.x) >> 5; // global wave id
    const int sub   = lane >> 3;          // which of the wave's 4 points
    const int cg4   = (lane & 7) * 4;     // channel offset (4 channels per lane)

    const int  p0    = wid * 4;           // first point of this wave
    const int  p     = p0 + sub;
    const bool valid = p < n;

    // ---- Stage this wave's 4-point coordinates into LDS via the async path.
    // Lanes 0-11 fetch pts[p0*3 .. p0*3+11], lanes 12-15 fetch tsv[p0..p0+3];
    // lanes 16-31 duplicate lanes 0-15 (idempotent writes keep EXEC full).
    {
        const int li = lane & 15;
        const float* gsrc;
        if (li < 12) {
            const int gi = min(p0 * 3 + li, n * 3 - 1);
            gsrc = pts + gi;
        } else {
            const int gi = min(p0 + (li - 12), n - 1);
            gsrc = tsv + gi;
        }
        const uint32_t laddr = (uint32_t)(uintptr_t)&csmem[wslot * 16 + li];
        asm volatile("global_load_async_to_lds_b32 %0, %1, off"
                     :: "v"(laddr), "v"((uint64_t)(uintptr_t)gsrc)
                     : "memory");
        asm volatile("s_wait_asynccnt 0x0" ::: "memory");
        // producer wave == consumer wave: no workgroup barrier needed
    }

    const float px = csmem[wslot * 16 + sub * 3 + 0];
    const float py = csmem[wslot * 16 + sub * 3 + 1];
    const float pz = csmem[wslot * 16 + sub * 3 + 2];
    const float tt = csmem[wslot * 16 + 12 + sub];

    // normalize_aabb with aabb=[[1.6]*3, [-1.6]*3]  (folds to c = -p/1.6)
    const float k  = 2.0f / (-2.0f * 1.6f);
    const float cx = (px - 1.6f) * k - 1.0f;
    const float cy = (py - 1.6f) * k - 1.0f;
    const float cz = (pz - 1.6f) * k - 1.0f;
    // normalize_time: 2*t*300/299 - 1
    const float ct = tt * (2.0f * 300.0f / 299.0f) - 1.0f;

    // Time quantization (reso 150) is identical across all levels.
    const Q qt = quant<150>(ct);

    const v4f f0 = level_feat<0>(pl, cx, cy, cz, qt, cg4);
    const v4f f1 = level_feat<1>(pl, cx, cy, cz, qt, cg4);
    const v4f f2 = level_feat<2>(pl, cx, cy, cz, qt, cg4);

    if (valid) {
        // Output is write-once streaming -> non-temporal b128 stores so the
        // 192 MB output stream does not evict the L2-resident plane data.
        float* o = out + (size_t)p * 96 + cg4;
        __builtin_nontemporal_store(f0, (v4f*)(o));
        __builtin_nontemporal_store(f1, (v4f*)(o + 32));
        __builtin_nontemporal_store(f2, (v4f*)(o + 64));
    }
}

extern "C" void kernel_launch(void* const* d_in, const int* in_sizes, int n_in,
                              void* d_out, int out_size, void* d_ws, size_t ws_size,
                              hipStream_t stream) {
    (void)n_in; (void)out_size; (void)d_ws; (void)ws_size;
    const float* pts = (const float*)d_in[0];
    const float* tsv = (const float*)d_in[1];
    // d_in[2] = scales (dead code in reference)
    PlanePtrs pl;
    for (int i = 0; i < 18; ++i) pl.p[i] = (const float*)d_in[3 + i];

    float* out = (float*)d_out;
    const int n = in_sizes[0] / 3;
    if (n <= 0) return;

    // 256 threads = 8 waves/block, 4 points/wave -> 32 points/block
    const int blocks = (n + 31) / 32;
    hexplane_kernel<<<blocks, 256, 0, stream>>>(pts, tsv, pl, out, n);
}